// DPCABlock_30477087932754
// MI455X (gfx1250) — compile-verified
//
#include <hip/hip_runtime.h>
#include <math.h>

// ---------------- problem constants (from reference) ----------------
#define Bq      4
#define CDIM    256
#define NPIX    4096            // H*W = 64*64
#define HEADS   8
#define DH      32
#define BHN     32              // B*HEADS
#define NPTS    (BHN*NPIX)      // 131072 flattened points
#define KC      512             // centroids == TOP_K
#define KM_ITERS 8

typedef float v2f __attribute__((ext_vector_type(2)));
typedef float v8f __attribute__((ext_vector_type(8)));

static __device__ __forceinline__ v8f vzero8() {
    v8f z = {0.f,0.f,0.f,0.f,0.f,0.f,0.f,0.f};
    return z;
}

// D = A(16x4 f32) * B(4x16 f32) + C(16x16 f32), full fp32 precision.
static __device__ __forceinline__ v8f wmma_f32(v2f a, v2f b, v8f c) {
    return __builtin_amdgcn_wmma_f32_16x16x4_f32(false, a, false, b, (short)0, c, false, false);
}

// ---------------- per-column mean / rstd for chan_norm (fused later) ----------------
__global__ void colstats_kernel(const float* __restrict__ x,
                                float* __restrict__ mean, float* __restrict__ rstd) {
    int idx = blockIdx.x * blockDim.x + threadIdx.x;      // over B*NPIX columns
    if (idx >= Bq * NPIX) return;
    int b = idx >> 12, p = idx & (NPIX - 1);
    const float* col = x + (size_t)b * CDIM * NPIX + p;
    float s = 0.f;
    #pragma unroll 8
    for (int c = 0; c < CDIM; c++) s += col[(size_t)c * NPIX];
    float m = s * (1.f / CDIM);
    float ss = 0.f;
    #pragma unroll 8
    for (int c = 0; c < CDIM; c++) { float d = col[(size_t)c * NPIX] - m; ss += d * d; }
    mean[idx] = m;
    rstd[idx] = rsqrtf(ss * (1.f / CDIM) + 1e-5f);
}

// ---------------- projection GEMM: Out[o,p] = sum_c W[o,c] * chan_norm(X)[c,p] -------
// Writes directly into (bh, p, d) layout.  o<256 -> out_k, o>=256 -> out_v (kv split).
__global__ __launch_bounds__(256)
void gemm_proj_kernel(const float* __restrict__ Wt,
                      const float* __restrict__ X,
                      const float* __restrict__ mean, const float* __restrict__ rstd,
                      const float* __restrict__ g, const float* __restrict__ bb,
                      float* __restrict__ out_k, float* __restrict__ out_v) {
    const int lane = threadIdx.x & 31, wave = threadIdx.x >> 5;
    const int half = lane >> 4, lm = lane & 15;
    const int b  = blockIdx.z;
    const int o0 = (blockIdx.y * 8 + wave) * 16;
    const int p0 = blockIdx.x * 64;
    const float* xb = X + (size_t)b * CDIM * NPIX;

    float mn[4], rs[4];
    #pragma unroll
    for (int j = 0; j < 4; j++) {
        int p = p0 + j * 16 + lm;
        mn[j] = mean[b * NPIX + p];
        rs[j] = rstd[b * NPIX + p];
    }
    v8f acc[4];
    #pragma unroll
    for (int j = 0; j < 4; j++) acc[j] = vzero8();

    const float* wrow = Wt + (size_t)(o0 + lm) * CDIM + 2 * half;
    for (int s = 0; s < 64; s++) {
        int c = 4 * s + 2 * half;
        v2f a; a.x = wrow[4 * s]; a.y = wrow[4 * s + 1];
        float g0 = g[c], g1 = g[c + 1], bc0 = bb[c], bc1 = bb[c + 1];
        const float* x0p = xb + (size_t)c * NPIX + p0 + lm;
        #pragma unroll
        for (int j = 0; j < 4; j++) {
            float x0 = x0p[j * 16];
            float x1 = x0p[NPIX + j * 16];
            v2f bf;
            bf.x = (x0 - mn[j]) * rs[j] * g0 + bc0;
            bf.y = (x1 - mn[j]) * rs[j] * g1 + bc1;
            acc[j] = wmma_f32(a, bf, acc[j]);
        }
    }
    #pragma unroll
    for (int j = 0; j < 4; j++) {
        int p = p0 + j * 16 + lm;
        #pragma unroll
        for (int r = 0; r < 8; r++) {
            int o = o0 + r + 8 * half;
            float val = acc[j][r];
            if (o < 256)
                out_k[(((size_t)(b * HEADS + (o >> 5))) * NPIX + p) * DH + (o & 31)] = val;
            else {
                int o2 = o - 256;
                out_v[(((size_t)(b * HEADS + (o2 >> 5))) * NPIX + p) * DH + (o2 & 31)] = val;
            }
        }
    }
}

// ---------------- row-wise L2 normalization over 32 dims ----------------
__global__ void l2norm_kernel(float* __restrict__ buf) {
    size_t row = (size_t)blockIdx.x * blockDim.x + threadIdx.x;
    if (row >= (size_t)NPTS) return;
    float* x = buf + row * DH;
    float ss = 0.f;
    #pragma unroll
    for (int d = 0; d < DH; d++) ss += x[d] * x[d];
    float inv = 1.f / fmaxf(sqrtf(ss), 1e-12f);
    #pragma unroll
    for (int d = 0; d < DH; d++) x[d] *= inv;
}

// ---------------- k-means ----------------
__global__ void init_cent_kernel(const float* __restrict__ q,
                                 float* __restrict__ cent, float* __restrict__ cn2) {
    int c = blockIdx.x * blockDim.x + threadIdx.x;
    if (c >= KC) return;
    float ss = 0.f;
    #pragma unroll
    for (int d = 0; d < DH; d++) { float v = q[(size_t)c * DH + d]; cent[c * DH + d] = v; ss += v * v; }
    cn2[c] = ss;
}

__global__ void zero_kernel(float* __restrict__ p, int n) {
    int i = blockIdx.x * blockDim.x + threadIdx.x;
    if (i < n) p[i] = 0.f;
}

// Per-wave: argmin over 512 centroids for 16 points via WMMA (d = |c|^2 - 2 x.c).
static __device__ __forceinline__ void assign16(const float* __restrict__ pts, int row0,
                                                const float* __restrict__ cent,
                                                const float* __restrict__ cn2,
                                                int lane, int* best) {
    const int half = lane >> 4, lm = lane & 15;
    v2f afr[8];
    const float* prow = pts + (size_t)(row0 + lm) * DH + 2 * half;
    #pragma unroll
    for (int s = 0; s < 8; s++) { afr[s].x = prow[4 * s]; afr[s].y = prow[4 * s + 1]; }

    float bd[8]; int bi[8];
    #pragma unroll
    for (int r = 0; r < 8; r++) { bd[r] = 3.4e38f; bi[r] = 0; }

    for (int ct = 0; ct < KC / 16; ct++) {
        v8f acc = vzero8();
        const float* crow = cent + (size_t)(ct * 16 + lm) * DH + 2 * half;
        #pragma unroll
        for (int s = 0; s < 8; s++) {
            v2f bf; bf.x = crow[4 * s]; bf.y = crow[4 * s + 1];
            acc = wmma_f32(afr[s], bf, acc);
        }
        float c2 = cn2[ct * 16 + lm];
        int   n  = ct * 16 + lm;
        #pragma unroll
        for (int r = 0; r < 8; r++) {
            float d = c2 - 2.f * acc[r];
            if (d < bd[r]) { bd[r] = d; bi[r] = n; }   // n monotonically increasing -> first min kept
        }
    }
    // reduce (min dist, lowest index on tie) across the 16 lanes of each half
    #pragma unroll
    for (int r = 0; r < 8; r++) {
        float d = bd[r]; int i = bi[r];
        #pragma unroll
        for (int m = 1; m < 16; m <<= 1) {
            float d2 = __shfl_xor(d, m);
            int   i2 = __shfl_xor(i, m);
            if (d2 < d || (d2 == d && i2 < i)) { d = d2; i = i2; }
        }
        best[r] = i;
    }
}

__global__ __launch_bounds__(256)
void kmeans_assign_kernel(const float* __restrict__ pts,
                          const float* __restrict__ cent, const float* __restrict__ cn2,
                          float* __restrict__ csum, float* __restrict__ ccnt) {
    const int lane = threadIdx.x & 31, wave = threadIdx.x >> 5;
    const int half = lane >> 4, lm = lane & 15;
    const int row0 = (blockIdx.x * 8 + wave) * 16;
    int best[8];
    assign16(pts, row0, cent, cn2, lane, best);
    #pragma unroll
    for (int r = 0; r < 8; r++) {
        int row = row0 + r + 8 * half;
        int a = best[r];
        float x0 = pts[(size_t)row * DH + lm];
        float x1 = pts[(size_t)row * DH + lm + 16];
        atomicAdd(&csum[a * DH + lm], x0);
        atomicAdd(&csum[a * DH + lm + 16], x1);
        if (lm == 0) atomicAdd(&ccnt[a], 1.f);
    }
}

__global__ void kmeans_update_kernel(float* __restrict__ cent, float* __restrict__ cn2,
                                     const float* __restrict__ csum, const float* __restrict__ ccnt) {
    int c = blockIdx.x * blockDim.x + threadIdx.x;
    if (c >= KC) return;
    float cnt = ccnt[c];
    float ss = 0.f;
    #pragma unroll
    for (int d = 0; d < DH; d++) {
        float v = (cnt > 0.f) ? (csum[c * DH + d] / fmaxf(cnt, 1.f)) : cent[c * DH + d];
        cent[c * DH + d] = v;
        ss += v * v;
    }
    cn2[c] = ss;
}

// assign k-points + L1 distance to assigned centroid
__global__ __launch_bounds__(256)
void assign_dist_kernel(const float* __restrict__ pts,
                        const float* __restrict__ cent, const float* __restrict__ cn2,
                        float* __restrict__ dist) {
    const int lane = threadIdx.x & 31, wave = threadIdx.x >> 5;
    const int half = lane >> 4, lm = lane & 15;
    const int row0 = (blockIdx.x * 8 + wave) * 16;
    int best[8];
    assign16(pts, row0, cent, cn2, lane, best);
    #pragma unroll
    for (int r = 0; r < 8; r++) {
        int row = row0 + r + 8 * half;
        int a = best[r];
        float x0 = pts[(size_t)row * DH + lm];
        float x1 = pts[(size_t)row * DH + lm + 16];
        float part = fabsf(cent[a * DH + lm] - x0) + fabsf(cent[a * DH + lm + 16] - x1);
        #pragma unroll
        for (int m = 1; m < 16; m <<= 1) part += __shfl_xor(part, m);
        if (lm == 0) dist[row] = part;
    }
}

// ---------------- top-512 per bh: bitonic sort of 4096 packed keys in LDS ----------
__global__ __launch_bounds__(1024)
void topk_kernel(const float* __restrict__ dist, int* __restrict__ idx) {
    __shared__ unsigned long long key[NPIX];     // 32 KB
    const int bh = blockIdx.x;
    const float* drow = dist + (size_t)bh * NPIX;
    for (int i = threadIdx.x; i < NPIX; i += 1024) {
        unsigned u = __float_as_uint(drow[i]);
        u = (u & 0x80000000u) ? ~u : (u | 0x80000000u);     // monotone ascending map
        key[i] = ((unsigned long long)(~u) << 32) | (unsigned)i;  // ascending key == desc value, asc index
    }
    __syncthreads();
    for (int k = 2; k <= NPIX; k <<= 1) {
        for (int j = k >> 1; j > 0; j >>= 1) {
            for (int i = threadIdx.x; i < NPIX; i += 1024) {
                int ixj = i ^ j;
                if (ixj > i) {
                    bool up = ((i & k) == 0);
                    unsigned long long a = key[i], b = key[ixj];
                    if ((a > b) == up) { key[i] = b; key[ixj] = a; }
                }
            }
            __syncthreads();
        }
    }
    for (int t = threadIdx.x; t < KC; t += 1024)
        idx[bh * KC + t] = (int)(key[t] & 0xffffffffu);
}

__global__ void gather_kernel(const float* __restrict__ kb, const float* __restrict__ vb,
                              const int* __restrict__ idx,
                              float* __restrict__ ksel, float* __restrict__ vsel) {
    int t = blockIdx.x * blockDim.x + threadIdx.x;          // BHN*KC*DH
    if (t >= BHN * KC * DH) return;
    int d = t & 31, j = (t >> 5) & (KC - 1), bh = t >> 14;
    int src = idx[bh * KC + j];
    size_t so = ((size_t)bh * NPIX + src) * DH + d;
    ksel[t] = kb[so];
    vsel[t] = vb[so];
}

// ---------------- attention over 512 selected keys ----------------
__global__ __launch_bounds__(128)
void attn_kernel(const float* __restrict__ qb,
                 const float* __restrict__ kselg, const float* __restrict__ vselg,
                 float* __restrict__ ao) {
    __shared__ float ks[KC * DH];        // 64 KB
    __shared__ float vs[KC * DH];        // 64 KB
    __shared__ float ps[4][16 * KC];     // 128 KB  (per-wave 16x512 score scratch)
    const int bh = blockIdx.y;
    const int lane = threadIdx.x & 31, wave = threadIdx.x >> 5;
    const int half = lane >> 4, lm = lane & 15;

    for (int i = threadIdx.x; i < KC * DH; i += 128) {
        ks[i] = kselg[(size_t)bh * KC * DH + i];
        vs[i] = vselg[(size_t)bh * KC * DH + i];
    }
    __syncthreads();

    const int q0 = blockIdx.x * 64 + wave * 16;
    v2f afr[8];
    const float* qrow = qb + ((size_t)bh * NPIX + q0 + lm) * DH + 2 * half;
    #pragma unroll
    for (int s = 0; s < 8; s++) { afr[s].x = qrow[4 * s]; afr[s].y = qrow[4 * s + 1]; }

    float* psw = ps[wave];
    // phase 1: sim = q . k^T
    for (int ct = 0; ct < KC / 16; ct++) {
        v8f acc = vzero8();
        const float* krow = &ks[(ct * 16 + lm) * DH + 2 * half];
        #pragma unroll
        for (int s = 0; s < 8; s++) {
            v2f bf; bf.x = krow[4 * s]; bf.y = krow[4 * s + 1];
            acc = wmma_f32(afr[s], bf, acc);
        }
        #pragma unroll
        for (int r = 0; r < 8; r++) psw[(r + 8 * half) * KC + ct * 16 + lm] = acc[r];
    }
    __syncthreads();

    // phase 2: softmax. lane handles row lm, its half of the 512 columns
    {
        float* prow = &psw[lm * KC + half * 256];
        float mx = -3.4e38f;
        for (int j = 0; j < 256; j += 4) {
            float4 v = *(const float4*)&prow[j];
            mx = fmaxf(mx, fmaxf(fmaxf(v.x, v.y), fmaxf(v.z, v.w)));
        }
        mx = fmaxf(mx, __shfl_xor(mx, 16));
        float sum = 0.f;
        for (int j = 0; j < 256; j += 4) {
            float4 v = *(float4*)&prow[j];
            v.x = expf(v.x - mx); v.y = expf(v.y - mx);
            v.z = expf(v.z - mx); v.w = expf(v.w - mx);
            *(float4*)&prow[j] = v;
            sum += v.x + v.y + v.z + v.w;
        }
        sum += __shfl_xor(sum, 16);
        float inv = 1.f / sum;
        for (int j = 0; j < 256; j += 4) {
            float4 v = *(float4*)&prow[j];
            v.x *= inv; v.y *= inv; v.z *= inv; v.w *= inv;
            *(float4*)&prow[j] = v;
        }
    }
    __syncthreads();

    // phase 3: out = probs . v   (two 16-wide d tiles)
    v8f oacc0 = vzero8(), oacc1 = vzero8();
    for (int s = 0; s < 128; s++) {
        int kj = 4 * s + 2 * half;
        v2f a; a.x = psw[lm * KC + kj]; a.y = psw[lm * KC + kj + 1];
        v2f b0, b1;
        b0.x = vs[kj * DH + lm];       b0.y = vs[(kj + 1) * DH + lm];
        b1.x = vs[kj * DH + 16 + lm];  b1.y = vs[(kj + 1) * DH + 16 + lm];
        oacc0 = wmma_f32(a, b0, oacc0);
        oacc1 = wmma_f32(a, b1, oacc1);
    }
    #pragma unroll
    for (int r = 0; r < 8; r++) {
        size_t base = ((size_t)bh * NPIX + q0 + r + 8 * half) * DH;
        ao[base + lm]      = oacc0[r];
        ao[base + 16 + lm] = oacc1[r];
    }
}

// ---------------- output projection: Out2[b,c,p] = sum_o w_out[c,o]*AO[o,p] --------
__global__ __launch_bounds__(256)
void gemm_out_kernel(const float* __restrict__ Wo,
                     const float* __restrict__ aob, float* __restrict__ out2) {
    const int lane = threadIdx.x & 31, wave = threadIdx.x >> 5;
    const int half = lane >> 4, lm = lane & 15;
    const int b  = blockIdx.z;
    const int c0 = (blockIdx.y * 8 + wave) * 16;
    const int p0 = blockIdx.x * 64;
    v8f acc[4];
    #pragma unroll
    for (int j = 0; j < 4; j++) acc[j] = vzero8();

    const float* wrow = Wo + (size_t)(c0 + lm) * CDIM + 2 * half;
    for (int s = 0; s < 64; s++) {
        int o = 4 * s + 2 * half;
        v2f a; a.x = wrow[4 * s]; a.y = wrow[4 * s + 1];
        int h = o >> 5, dd = o & 31;                 // o even -> pair stays inside one head
        const float* ap = aob + ((size_t)(b * HEADS + h) * NPIX) * DH + dd;
        #pragma unroll
        for (int j = 0; j < 4; j++) {
            int p = p0 + j * 16 + lm;
            v2f bf; bf.x = ap[(size_t)p * DH]; bf.y = ap[(size_t)p * DH + 1];
            acc[j] = wmma_f32(a, bf, acc[j]);
        }
    }
    #pragma unroll
    for (int j = 0; j < 4; j++) {
        int p = p0 + j * 16 + lm;
        #pragma unroll
        for (int r = 0; r < 8; r++) {
            int cc = c0 + r + 8 * half;
            out2[((size_t)b * CDIM + cc) * NPIX + p] = acc[j][r];
        }
    }
}

// ---------------- final chan_norm + gamma*x + residual ----------------
__global__ void final_norm_kernel(const float* __restrict__ out2,
                                  const float* __restrict__ g, const float* __restrict__ bb,
                                  const float* __restrict__ gamma_p,
                                  const float* __restrict__ qsrc,
                                  float* __restrict__ out) {
    int idx = blockIdx.x * blockDim.x + threadIdx.x;
    if (idx >= Bq * NPIX) return;
    int b = idx >> 12, p = idx & (NPIX - 1);
    const float* col = out2 + (size_t)b * CDIM * NPIX + p;
    float s = 0.f;
    #pragma unroll 8
    for (int c = 0; c < CDIM; c++) s += col[(size_t)c * NPIX];
    float m = s * (1.f / CDIM);
    float ss = 0.f;
    #pragma unroll 8
    for (int c = 0; c < CDIM; c++) { float d = col[(size_t)c * NPIX] - m; ss += d * d; }
    float rstd = rsqrtf(ss * (1.f / CDIM) + 1e-5f);
    float gm = gamma_p[0];
    for (int c = 0; c < CDIM; c++) {
        float v = (col[(size_t)c * NPIX] - m) * rstd * g[c] + bb[c];
        size_t o = ((size_t)b * CDIM + c) * NPIX + p;
        out[o] = gm * v + qsrc[o];
    }
}

// ---------------- host-side launch ----------------
extern "C" void kernel_launch(void* const* d_in, const int* in_sizes, int n_in,
                              void* d_out, int out_size, void* d_ws, size_t ws_size,
                              hipStream_t stream) {
    (void)in_sizes; (void)n_in; (void)out_size; (void)ws_size;
    const float* qsrc  = (const float*)d_in[0];
    const float* ctx   = (const float*)d_in[1];
    const float* g_ctx = (const float*)d_in[2];
    const float* b_ctx = (const float*)d_in[3];
    const float* g_qs  = (const float*)d_in[4];
    const float* b_qs  = (const float*)d_in[5];
    const float* g_out = (const float*)d_in[6];
    const float* b_out = (const float*)d_in[7];
    const float* w_q   = (const float*)d_in[8];
    const float* w_kv  = (const float*)d_in[9];
    const float* w_out = (const float*)d_in[10];
    const float* gamma = (const float*)d_in[11];
    float* out = (float*)d_out;

    float* ws = (float*)d_ws;
    size_t off = 0;
    auto alloc = [&](size_t n) { float* p = ws + off; off += n; return p; };
    float* mean_ctx = alloc(Bq * NPIX);
    float* rstd_ctx = alloc(Bq * NPIX);
    float* mean_qs  = alloc(Bq * NPIX);
    float* rstd_qs  = alloc(Bq * NPIX);
    float* q_buf = alloc((size_t)NPTS * DH);      // 16 MB, (bh,p,d)
    float* k_buf = alloc((size_t)NPTS * DH);      // 16 MB
    float* v_buf = alloc((size_t)NPTS * DH);      // 16 MB
    float* cent  = alloc(KC * DH);
    float* cn2   = alloc(KC);
    float* csum  = alloc(KC * DH);                // csum & ccnt contiguous (zeroed together)
    float* ccnt  = alloc(KC);
    float* dist  = alloc(NPTS);
    int*   tidx  = (int*)alloc(BHN * KC);
    float* ksel  = alloc((size_t)BHN * KC * DH);  // 2 MB
    float* vsel  = alloc((size_t)BHN * KC * DH);  // 2 MB
    float* ao    = alloc((size_t)NPTS * DH);      // 16 MB
    float* out2  = q_buf;                         // q dead after attention: reuse

    colstats_kernel<<<64, 256, 0, stream>>>(ctx,  mean_ctx, rstd_ctx);
    colstats_kernel<<<64, 256, 0, stream>>>(qsrc, mean_qs,  rstd_qs);

    gemm_proj_kernel<<<dim3(64, 2, Bq), 256, 0, stream>>>(w_q,  qsrc, mean_qs,  rstd_qs,
                                                          g_qs,  b_qs,  q_buf, q_buf);
    gemm_proj_kernel<<<dim3(64, 4, Bq), 256, 0, stream>>>(w_kv, ctx,  mean_ctx, rstd_ctx,
                                                          g_ctx, b_ctx, k_buf, v_buf);
    l2norm_kernel<<<NPTS / 256, 256, 0, stream>>>(q_buf);
    l2norm_kernel<<<NPTS / 256, 256, 0, stream>>>(k_buf);

    init_cent_kernel<<<2, 256, 0, stream>>>(q_buf, cent, cn2);
    for (int it = 0; it < KM_ITERS; ++it) {
        zero_kernel<<<(KC * DH + KC + 255) / 256, 256, 0, stream>>>(csum, KC * DH + KC);
        kmeans_assign_kernel<<<NPTS / 16 / 8, 256, 0, stream>>>(q_buf, cent, cn2, csum, ccnt);
        kmeans_update_kernel<<<2, 256, 0, stream>>>(cent, cn2, csum, ccnt);
    }
    assign_dist_kernel<<<NPTS / 16 / 8, 256, 0, stream>>>(k_buf, cent, cn2, dist);

    topk_kernel<<<BHN, 1024, 0, stream>>>(dist, tidx);
    gather_kernel<<<(BHN * KC * DH) / 256, 256, 0, stream>>>(k_buf, v_buf, tidx, ksel, vsel);

    attn_kernel<<<dim3(64, BHN), 128, 0, stream>>>(q_buf, ksel, vsel, ao);

    gemm_out_kernel<<<dim3(64, 2, Bq), 256, 0, stream>>>(w_out, ao, out2);
    final_norm_kernel<<<64, 256, 0, stream>>>(out2, g_out, b_out, gamma, qsrc, out);
}